// ContactMapLinear_70995809403345
// MI455X (gfx1250) — compile-verified
//
#include <hip/hip_runtime.h>
#include <stdint.h>

typedef __attribute__((ext_vector_type(2))) float v2f;
typedef __attribute__((ext_vector_type(8))) float v8f;

#define D_DIM 4096
#define I_DIM 1024

// Block tile: 128(M) x 64(N), K staged in chunks of 32, double-buffered LDS.
#define BM 128
#define BN 64
#define BK 32
#define AST 36                             // padded LDS row stride (floats): 16B-aligned rows, conflict-free b64 reads
#define A_FLOATS (BM * AST)                // 4608
#define B_FLOATS (BN * AST)                // 2304
#define BUF_FLOATS (A_FLOATS + B_FLOATS)   // 6912 floats -> 2 bufs = 55296 B LDS

// CDNA5 async global->LDS copy (ASYNCcnt); inline asm is portable across
// ROCm-7.2 and amdgpu-toolchain (the clang builtin arity differs between them).
__device__ __forceinline__ void async_copy_b128(uint32_t lds_byte_addr, const float* gaddr) {
  asm volatile("global_load_async_to_lds_b128 %0, %1, off"
               :: "v"(lds_byte_addr), "v"(gaddr)
               : "memory");
}
__device__ __forceinline__ void wait_async0() {
  asm volatile("s_wait_asynccnt 0" ::: "memory");
}

// D(MxN) = A(MxK) * B^T (B stored N x K row-major), optional strict-lower-tri mask.
// 256 threads = 8 wave32; each wave owns a 32x32 output tile (2x2 WMMA tiles).
__global__ __launch_bounds__(256) void wmma_gemm_nt(
    const float* __restrict__ A,
    const float* __restrict__ B,
    float* __restrict__ D,
    int K, int lda, int ldb, int ldd, int tril)
{
  __shared__ __align__(16) float sm[2][BUF_FLOATS];

  const int lane = threadIdx.x & 31;
  const int wave = threadIdx.y;            // 0..7
  const int tid  = wave * 32 + lane;
  const int wm   = wave & 3;               // 4 waves along M
  const int wn   = wave >> 2;              // 2 waves along N
  const int rb   = blockIdx.y * BM;
  const int cb   = blockIdx.x * BN;
  const int m0   = rb + wm * 32;
  const int n0   = cb + wn * 32;
  const int half = lane >> 4;              // lanes 0-15: K {0,1}; 16-31: K {2,3}
  const int l16  = lane & 15;

  v8f acc00 = {}, acc01 = {}, acc10 = {}, acc11 = {};

  // Block-uniform skip: tile region has a row>col element iff rb+BM-1 > cb.
  const bool skip = tril && (cb >= rb + BM - 1);

  if (!skip) {
    const float* Ag = A + (size_t)rb * lda;
    const float* Bg = B + (size_t)cb * ldb;
    const uint32_t lds0 = (uint32_t)(uintptr_t)(&sm[0][0]);

    // Stage one (BM+BN) x BK f32 tile into LDS buffer `buf` with async b128 copies.
    // A: 128 rows * 8 chunks(16B) = 1024 chunks (4/thread); B: 64 rows -> 512 (2/thread).
    auto stage = [&](int buf, int k0) {
      const uint32_t la = lds0 + (uint32_t)buf * (BUF_FLOATS * 4u);
      const uint32_t lb = la + A_FLOATS * 4u;
      #pragma unroll
      for (int i = 0; i < 4; ++i) {
        const int c = tid + 256 * i;
        const int r = c >> 3, q = c & 7;
        async_copy_b128(la + (uint32_t)(r * AST + q * 4) * 4u,
                        Ag + (size_t)r * lda + k0 + q * 4);
      }
      #pragma unroll
      for (int i = 0; i < 2; ++i) {
        const int c = tid + 256 * i;
        const int r = c >> 3, q = c & 7;
        async_copy_b128(lb + (uint32_t)(r * AST + q * 4) * 4u,
                        Bg + (size_t)r * ldb + k0 + q * 4);
      }
    };

    const int nkt = K / BK;
    stage(0, 0);
    for (int kt = 0; kt < nkt; ++kt) {
      wait_async0();          // this wave's copies into sm[kt&1] have landed
      __syncthreads();        // everyone's copies have landed
      if (kt + 1 < nkt) stage((kt + 1) & 1, (kt + 1) * BK);  // overlap next copy with compute

      const float* sA  = &sm[kt & 1][0]        + (wm * 32) * AST;
      const float* sB  = &sm[kt & 1][A_FLOATS] + (wn * 32) * AST;
      const float* a0p = sA + (l16)      * AST + 2 * half;
      const float* a1p = sA + (16 + l16) * AST + 2 * half;
      const float* b0p = sB + (l16)      * AST + 2 * half;
      const float* b1p = sB + (16 + l16) * AST + 2 * half;
      #pragma unroll
      for (int kk = 0; kk < BK; kk += 4) {
        v2f a0 = *(const v2f*)(a0p + kk);
        v2f a1 = *(const v2f*)(a1p + kk);
        v2f b0 = *(const v2f*)(b0p + kk);
        v2f b1 = *(const v2f*)(b1p + kk);
        acc00 = __builtin_amdgcn_wmma_f32_16x16x4_f32(false, a0, false, b0, (short)0, acc00, false, false);
        acc01 = __builtin_amdgcn_wmma_f32_16x16x4_f32(false, a0, false, b1, (short)0, acc01, false, false);
        acc10 = __builtin_amdgcn_wmma_f32_16x16x4_f32(false, a1, false, b0, (short)0, acc10, false, false);
        acc11 = __builtin_amdgcn_wmma_f32_16x16x4_f32(false, a1, false, b1, (short)0, acc11, false, false);
      }
      __syncthreads();        // all waves done reading sm[kt&1] before it is overwritten
    }
  }

  // C/D 16x16 f32 layout: VGPR r -> row r + 8*half, col = l16.
  const v8f accs[4] = {acc00, acc01, acc10, acc11};
  #pragma unroll
  for (int t = 0; t < 4; ++t) {
    const int mt = t >> 1;
    const int nt = t & 1;
    #pragma unroll
    for (int r = 0; r < 8; ++r) {
      const int row = m0 + 16 * mt + r + 8 * half;
      const int col = n0 + 16 * nt + l16;
      float v = accs[t][r];
      if (tril) v = (row > col) ? v : 0.0f;
      D[(size_t)row * ldd + col] = v;
    }
  }
}

// 32x32 LDS transpose: dst(cols x rows) = src(rows x cols)^T
__global__ __launch_bounds__(256) void transpose_f32(
    const float* __restrict__ src, float* __restrict__ dst, int rows, int cols)
{
  __shared__ float t[32][33];
  const int c0 = blockIdx.x * 32;
  const int r0 = blockIdx.y * 32;
  #pragma unroll
  for (int i = threadIdx.y; i < 32; i += 8)
    t[i][threadIdx.x] = src[(size_t)(r0 + i) * cols + c0 + threadIdx.x];
  __syncthreads();
  #pragma unroll
  for (int i = threadIdx.y; i < 32; i += 8)
    dst[(size_t)(c0 + i) * rows + r0 + threadIdx.x] = t[threadIdx.x][i];
}

extern "C" void kernel_launch(void* const* d_in, const int* in_sizes, int n_in,
                              void* d_out, int out_size, void* d_ws, size_t ws_size,
                              hipStream_t stream) {
  const float* features = (const float*)d_in[0];  // (1, 4097, 4096) f32
  const float* P        = (const float*)d_in[1];  // (4096, 1024) f32
  const float* Q        = (const float*)d_in[2];  // (1024, 4096) f32
  float* out            = (float*)d_out;          // (4096, 4096) f32

  const float* X = features + D_DIM;              // features[0, 1:, :], 4096x4096, ld=4096

  // Workspace: Pt (1024x4096) + A (4096x1024) + C (4096x1024) = 48 MB
  float* Pt = (float*)d_ws;
  float* Aw = Pt + (size_t)I_DIM * D_DIM;
  float* Cw = Aw + (size_t)D_DIM * I_DIM;

  dim3 blk(32, 8);

  // P (4096x1024) -> Pt (1024x4096): makes stage 1 an NT GEMM too.
  transpose_f32<<<dim3(I_DIM / 32, D_DIM / 32), blk, 0, stream>>>(P, Pt, D_DIM, I_DIM);

  // Stage 1: Aw = X @ P = X @ Pt^T     (M=4096, N=1024, K=4096)
  wmma_gemm_nt<<<dim3(I_DIM / BN, D_DIM / BM), blk, 0, stream>>>(
      X, Pt, Aw, D_DIM, D_DIM, D_DIM, I_DIM, 0);

  // Stage 2: Cw = X @ Q^T              (M=4096, N=1024, K=4096)
  wmma_gemm_nt<<<dim3(I_DIM / BN, D_DIM / BM), blk, 0, stream>>>(
      X, Q, Cw, D_DIM, D_DIM, D_DIM, I_DIM, 0);

  // Stage 3: out = tril(Aw @ Cw^T, -1) (M=N=4096, K=1024)
  wmma_gemm_nt<<<dim3(D_DIM / BN, D_DIM / BM), blk, 0, stream>>>(
      Aw, Cw, out, I_DIM, I_DIM, I_DIM, D_DIM, 1);
}